// PretrainedBlocks_14242111553628
// MI455X (gfx1250) — compile-verified
//
#include <hip/hip_runtime.h>
#include <math.h>

typedef __attribute__((ext_vector_type(2))) float v2f;
typedef __attribute__((ext_vector_type(8))) float v8f;

#define HEADS 4
#define EPS   1e-5f

// ---------------------------------------------------------------------------
// Utility kernels
// ---------------------------------------------------------------------------
__global__ void set_val_kernel(float* __restrict__ p, float v, long long n) {
    long long i = (long long)blockIdx.x * blockDim.x + threadIdx.x;
    if (i < n) p[i] = v;
}

__global__ void copy_kernel(const float* __restrict__ src, float* __restrict__ dst, long long n) {
    long long i = (long long)blockIdx.x * blockDim.x + threadIdx.x;
    if (i < n) dst[i] = src[i];
}

// deg starts at 1.0 (self loop); add 1 per incoming edge
__global__ void deg_edge_kernel(const int* __restrict__ eidx, int E, float* __restrict__ deg) {
    int e = blockIdx.x * blockDim.x + threadIdx.x;
    if (e < E) {
        int d = eidx[E + e];
        __hip_atomic_fetch_add(&deg[d], 1.0f, __ATOMIC_RELAXED, __HIP_MEMORY_SCOPE_AGENT);
    }
}

__global__ void inv_deg_kernel(float* __restrict__ deg, int nN) {
    int i = blockIdx.x * blockDim.x + threadIdx.x;
    if (i < nN) deg[i] = 1.0f / fmaxf(deg[i], 1.0f);
}

// ---------------------------------------------------------------------------
// Node transform: [Wx | Ux] = x @ [W | U] via V_WMMA_F32_16X16X4_F32.
//   x:   [Npad, FIN] (pad rows zeroed)  W: [FIN, 64]  U: [FIN, 4]
//   Wx:  [Npad, 64]  Ux16: [Npad, 16] (cols 4..15 are exact zeros)
// One wave32 per 16-node tile; node dim padded -> NO guards, NO exec churn.
// A (16x4 f32): lanes 0-15 -> M=lane, {v0:K0, v1:K1};
//               lanes 16-31 -> M=lane-16, {v0:K2, v1:K3}.
// B (4x16 f32): lanes 0-15 -> N=lane, {v0:K0, v1:K1};
//               lanes 16-31 -> N=lane-16, {v0:K2, v1:K3}.
// C/D (16x16 f32): vgpr g, lanes 0-15 -> D[g][lane]; lanes 16-31 -> D[g+8][lane-16].
// ---------------------------------------------------------------------------
template <int FIN>
__global__ void node_transform_wmma(const float* __restrict__ xin,
                                    const float* __restrict__ W,
                                    const float* __restrict__ U,
                                    float* __restrict__ Wx,
                                    float* __restrict__ Ux16,
                                    int nTiles) {
    const int lane = threadIdx.x & 31;
    const int tile = blockIdx.x * (blockDim.x >> 5) + (threadIdx.x >> 5);
    if (tile >= nTiles) return;  // wave-uniform

    const int row0  = tile * 16;
    const int khalf = lane >> 4;       // 0 or 1
    const int coln  = lane & 15;
    const int node  = row0 + coln;     // A row for this lane (pad rows valid)

    constexpr int NK = FIN / 4;
    v2f a[NK];
#pragma unroll
    for (int k = 0; k < NK; ++k) {
        int c0 = 4 * k + khalf * 2;
        a[k].x = xin[(long long)node * FIN + c0];
        a[k].y = xin[(long long)node * FIN + c0 + 1];
    }

#pragma unroll
    for (int nt = 0; nt < 5; ++nt) {
        v8f acc = {};
#pragma unroll
        for (int k = 0; k < NK; ++k) {
            int kk = 4 * k + khalf * 2;
            v2f bfrag;
            if (nt < 4) {
                int gc = nt * 16 + coln;
                bfrag.x = W[kk * 64 + gc];
                bfrag.y = W[(kk + 1) * 64 + gc];
            } else {
                bfrag.x = (coln < HEADS) ? U[kk * HEADS + coln] : 0.0f;
                bfrag.y = (coln < HEADS) ? U[(kk + 1) * HEADS + coln] : 0.0f;
            }
            acc = __builtin_amdgcn_wmma_f32_16x16x4_f32(
                false, a[k], false, bfrag, (short)0, acc, false, false);
        }

        if (nt < 4) {
            float* dst = Wx + (long long)(row0 + khalf * 8) * 64 + nt * 16 + coln;
#pragma unroll
            for (int g = 0; g < 8; ++g) dst[g * 64] = acc[g];      // unconditional
        } else {
            float* dst = Ux16 + (long long)(row0 + khalf * 8) * 16 + coln;
#pragma unroll
            for (int g = 0; g < 8; ++g) dst[g * 16] = acc[g];      // cols 4..15 == 0
        }
    }
}

// ---------------------------------------------------------------------------
// Edge kernel: attention softmax over 4 heads + scatter-add of messages.
// Edges e < E come from edge_index; e >= E are self loops (s = d = e - E).
// Wx row read as float4 -> global_load_b128 (dominant traffic, L2-resident).
// ---------------------------------------------------------------------------
__global__ void feast_edge_kernel(const int* __restrict__ eidx, int E, int nN,
                                  const float* __restrict__ Ux16,
                                  const float* __restrict__ Wx,
                                  const float* __restrict__ cvec,
                                  float* __restrict__ agg) {
    long long e = (long long)blockIdx.x * blockDim.x + threadIdx.x;
    long long total = (long long)E + nN;
    if (e >= total) return;

    int s, d;
    if (e < E) { s = eidx[e]; d = eidx[(long long)E + e]; }
    else       { s = d = (int)(e - E); }

    float4 uS = *(const float4*)(Ux16 + (long long)s * 16);
    float4 uD = *(const float4*)(Ux16 + (long long)d * 16);
    float u0 = uS.x - uD.x + cvec[0];
    float u1 = uS.y - uD.y + cvec[1];
    float u2 = uS.z - uD.z + cvec[2];
    float u3 = uS.w - uD.w + cvec[3];

    float m  = fmaxf(fmaxf(u0, u1), fmaxf(u2, u3));
    float e0 = __expf(u0 - m), e1 = __expf(u1 - m), e2 = __expf(u2 - m), e3 = __expf(u3 - m);
    float inv = 1.0f / (e0 + e1 + e2 + e3);
    float q0 = e0 * inv, q1 = e1 * inv, q2 = e2 * inv, q3 = e3 * inv;

    const float4* w4 = (const float4*)(Wx + (long long)s * 64);  // 16 x float4
    float* ad = agg + (long long)d * 16;
#pragma unroll
    for (int c4 = 0; c4 < 4; ++c4) {     // 4 output columns per chunk
        float4 a0 = w4[c4];              // head 0
        float4 a1 = w4[4 + c4];          // head 1
        float4 a2 = w4[8 + c4];          // head 2
        float4 a3 = w4[12 + c4];         // head 3
        float m0 = q0 * a0.x + q1 * a1.x + q2 * a2.x + q3 * a3.x;
        float m1 = q0 * a0.y + q1 * a1.y + q2 * a2.y + q3 * a3.y;
        float m2 = q0 * a0.z + q1 * a1.z + q2 * a2.z + q3 * a3.z;
        float m3 = q0 * a0.w + q1 * a1.w + q2 * a2.w + q3 * a3.w;
        __hip_atomic_fetch_add(&ad[c4 * 4 + 0], m0, __ATOMIC_RELAXED, __HIP_MEMORY_SCOPE_AGENT);
        __hip_atomic_fetch_add(&ad[c4 * 4 + 1], m1, __ATOMIC_RELAXED, __HIP_MEMORY_SCOPE_AGENT);
        __hip_atomic_fetch_add(&ad[c4 * 4 + 2], m2, __ATOMIC_RELAXED, __HIP_MEMORY_SCOPE_AGENT);
        __hip_atomic_fetch_add(&ad[c4 * 4 + 3], m3, __ATOMIC_RELAXED, __HIP_MEMORY_SCOPE_AGENT);
    }
}

// out = relu(agg * inv_deg + bias)
template <int FOUT>
__global__ void conv_finalize_kernel(const float* __restrict__ agg,
                                     const float* __restrict__ inv_deg,
                                     const float* __restrict__ bias,
                                     float* __restrict__ out, int nN) {
    long long i = (long long)blockIdx.x * blockDim.x + threadIdx.x;
    if (i < (long long)nN * FOUT) {
        int node = (int)(i / FOUT);
        int o    = (int)(i % FOUT);
        out[i] = fmaxf(agg[i] * inv_deg[node] + bias[o], 0.0f);
    }
}

// ---------------------------------------------------------------------------
// MLP tail per node: h16 -> relu(lin1) -> lin2 -> (out head, residual sigmoid)
// Weights staged through LDS once per block.
// ---------------------------------------------------------------------------
__global__ void mlp_kernel(const float* __restrict__ h16,
                           const float* __restrict__ W1, const float* __restrict__ b1,
                           const float* __restrict__ W2, const float* __restrict__ b2,
                           const float* __restrict__ W3, const float* __restrict__ b3,
                           float* __restrict__ x_cur, float* __restrict__ out, int nN) {
    __shared__ float sW1[16 * 32], sb1[32], sW2[32 * 4], sb2[4], sW3[4], sb3[1];
    for (int t = threadIdx.x; t < 16 * 32; t += blockDim.x) sW1[t] = W1[t];
    for (int t = threadIdx.x; t < 32;      t += blockDim.x) sb1[t] = b1[t];
    for (int t = threadIdx.x; t < 32 * 4;  t += blockDim.x) sW2[t] = W2[t];
    if (threadIdx.x < 4) { sb2[threadIdx.x] = b2[threadIdx.x]; sW3[threadIdx.x] = W3[threadIdx.x]; }
    if (threadIdx.x == 0) sb3[0] = b3[0];
    __syncthreads();

    int i = blockIdx.x * blockDim.x + threadIdx.x;
    if (i >= nN) return;

    float h[16];
#pragma unroll
    for (int k = 0; k < 16; ++k) h[k] = h16[(long long)i * 16 + k];

    float inter[4] = {sb2[0], sb2[1], sb2[2], sb2[3]};
#pragma unroll
    for (int j = 0; j < 32; ++j) {
        float g = sb1[j];
#pragma unroll
        for (int k = 0; k < 16; ++k) g += h[k] * sW1[k * 32 + j];
        g = fmaxf(g, 0.0f);
#pragma unroll
        for (int c = 0; c < 4; ++c) inter[c] += g * sW2[j * 4 + c];
    }

    float o = sb3[0];
#pragma unroll
    for (int c = 0; c < 4; ++c) o += fmaxf(inter[c], 0.0f) * sW3[c];
    out[i] = 1.0f / (1.0f + __expf(-o));

#pragma unroll
    for (int c = 0; c < 4; ++c)
        x_cur[(long long)i * 4 + c] += 1.0f / (1.0f + __expf(-inter[c]));
}

// ---------------------------------------------------------------------------
// BatchNorm over the 4-wide residual stream: two-pass (sum & sumsq -> apply)
// ---------------------------------------------------------------------------
__global__ void bn_reduce_kernel(const float* __restrict__ x, float* __restrict__ red, int nN) {
    __shared__ float s[8];
    if (threadIdx.x < 8) s[threadIdx.x] = 0.0f;
    __syncthreads();
    int i = blockIdx.x * blockDim.x + threadIdx.x;
    if (i < nN) {
#pragma unroll
        for (int c = 0; c < 4; ++c) {
            float v = x[(long long)i * 4 + c];
            atomicAdd(&s[c], v);
            atomicAdd(&s[4 + c], v * v);
        }
    }
    __syncthreads();
    if (threadIdx.x < 8)
        __hip_atomic_fetch_add(&red[threadIdx.x], s[threadIdx.x],
                               __ATOMIC_RELAXED, __HIP_MEMORY_SCOPE_AGENT);
}

__global__ void bn_apply_kernel(float* __restrict__ x, const float* __restrict__ red,
                                const float* __restrict__ gamma, const float* __restrict__ beta,
                                int nN) {
    long long i = (long long)blockIdx.x * blockDim.x + threadIdx.x;
    if (i < (long long)nN * 4) {
        int c = (int)(i & 3);
        float invN = 1.0f / (float)nN;
        float mu   = red[c] * invN;
        float var  = red[4 + c] * invN - mu * mu;
        x[i] = (x[i] - mu) * rsqrtf(var + EPS) * gamma[c] + beta[c];
    }
}

// ---------------------------------------------------------------------------
// Host orchestration
// ---------------------------------------------------------------------------
static inline int cdiv_ll(long long a, int b) { return (int)((a + b - 1) / b); }

extern "C" void kernel_launch(void* const* d_in, const int* in_sizes, int n_in,
                              void* d_out, int out_size, void* d_ws, size_t ws_size,
                              hipStream_t stream) {
    const float* x_in = (const float*)d_in[0];
    const int*   eidx = (const int*)d_in[1];
    const int N    = in_sizes[0] / 4;
    const int E    = in_sizes[1] / 2;
    const int Npad = (N + 15) & ~15;

    auto P = [&](int i) { return (const float*)d_in[i]; };
    // Param flatten order (JAX sorted-key tree): blocks[0..2] then bns[0..1].
    // Per block (18 leaves): conv{1,2,3}:{U,W,b,c}, lin1:{W,b}, lin2:{W,b}, out:{W,b}.
    // Per bn (2 leaves): {beta, gamma}.

    // Workspace layout (floats); node-dim buffers padded to Npad rows.
    float* ws = (float*)d_ws;
    size_t off = 0;
    float* x_cur = ws + off; off += (size_t)Npad * 4;
    float* dinv  = ws + off; off += (size_t)Npad;       // deg, then inverted in place
    float* Ux16  = ws + off; off += (size_t)Npad * 16;  // stride-16, cols 4..15 zero
    float* Wx    = ws + off; off += (size_t)Npad * 64;
    float* agg   = ws + off; off += (size_t)Npad * 16;
    float* hA    = ws + off; off += (size_t)Npad * 16;
    float* hB    = ws + off; off += (size_t)Npad * 16;
    float* red   = ws + off; off += 8;

    float* out = (float*)d_out;
    const int B256 = 256;

    // residual stream init + zero pad rows of all WMMA input buffers
    copy_kernel<<<cdiv_ll((long long)N * 4, B256), B256, 0, stream>>>(x_in, x_cur, (long long)N * 4);
    if (Npad > N) {
        int padN = Npad - N;
        set_val_kernel<<<1, 64, 0, stream>>>(x_cur + (long long)N * 4, 0.0f, (long long)padN * 4);
        set_val_kernel<<<1, 256, 0, stream>>>(hA + (long long)N * 16, 0.0f, (long long)padN * 16);
        set_val_kernel<<<1, 256, 0, stream>>>(hB + (long long)N * 16, 0.0f, (long long)padN * 16);
    }

    // degrees (self loop contributes the initial 1.0)
    set_val_kernel<<<cdiv_ll(N, B256), B256, 0, stream>>>(dinv, 1.0f, N);
    deg_edge_kernel<<<cdiv_ll(E, B256), B256, 0, stream>>>(eidx, E, dinv);
    inv_deg_kernel<<<cdiv_ll(N, B256), B256, 0, stream>>>(dinv, N);

    const int nTiles   = Npad / 16;
    const int wmmaGrid = (nTiles + 7) / 8;  // 8 waves / 256-thread block
    const long long totE = (long long)E + N;

    for (int b = 0; b < 3; ++b) {
        const int pb = 2 + b * 18;

        // ---- conv1 (fin=4), input = x_cur ----
        node_transform_wmma<4><<<wmmaGrid, 256, 0, stream>>>(
            x_cur, P(pb + 1), P(pb + 0), Wx, Ux16, nTiles);
        set_val_kernel<<<cdiv_ll((long long)N * 16, B256), B256, 0, stream>>>(agg, 0.0f, (long long)N * 16);
        feast_edge_kernel<<<cdiv_ll(totE, B256), B256, 0, stream>>>(
            eidx, E, N, Ux16, Wx, P(pb + 3), agg);
        conv_finalize_kernel<16><<<cdiv_ll((long long)N * 16, B256), B256, 0, stream>>>(
            agg, dinv, P(pb + 2), hA, N);

        // ---- conv2 (fin=16), input = hA ----
        node_transform_wmma<16><<<wmmaGrid, 256, 0, stream>>>(
            hA, P(pb + 5), P(pb + 4), Wx, Ux16, nTiles);
        set_val_kernel<<<cdiv_ll((long long)N * 16, B256), B256, 0, stream>>>(agg, 0.0f, (long long)N * 16);
        feast_edge_kernel<<<cdiv_ll(totE, B256), B256, 0, stream>>>(
            eidx, E, N, Ux16, Wx, P(pb + 7), agg);
        conv_finalize_kernel<16><<<cdiv_ll((long long)N * 16, B256), B256, 0, stream>>>(
            agg, dinv, P(pb + 6), hB, N);

        // ---- conv3 (fin=16), input = hB ----
        node_transform_wmma<16><<<wmmaGrid, 256, 0, stream>>>(
            hB, P(pb + 9), P(pb + 8), Wx, Ux16, nTiles);
        set_val_kernel<<<cdiv_ll((long long)N * 16, B256), B256, 0, stream>>>(agg, 0.0f, (long long)N * 16);
        feast_edge_kernel<<<cdiv_ll(totE, B256), B256, 0, stream>>>(
            eidx, E, N, Ux16, Wx, P(pb + 11), agg);
        conv_finalize_kernel<16><<<cdiv_ll((long long)N * 16, B256), B256, 0, stream>>>(
            agg, dinv, P(pb + 10), hA, N);

        // ---- MLP tail + residual + output head ----
        mlp_kernel<<<cdiv_ll(N, B256), B256, 0, stream>>>(
            hA, P(pb + 12), P(pb + 13), P(pb + 14), P(pb + 15), P(pb + 16), P(pb + 17),
            x_cur, out, N);

        // ---- BatchNorm between blocks ----
        if (b < 2) {
            const int pbn = 2 + 54 + b * 2;  // {beta, gamma}
            set_val_kernel<<<1, 8, 0, stream>>>(red, 0.0f, 8);
            bn_reduce_kernel<<<cdiv_ll(N, B256), B256, 0, stream>>>(x_cur, red, N);
            bn_apply_kernel<<<cdiv_ll((long long)N * 4, B256), B256, 0, stream>>>(
                x_cur, red, P(pbn + 1), P(pbn + 0), N);
        }
    }
}